// Embedding_18305150615599
// MI455X (gfx1250) — compile-verified
//
#include <hip/hip_runtime.h>
#include <hip/hip_bf16.h>

// Embedding gather: out[row, :] = weight[token_ids[row], :]
//   rows = BATCH*SEQ = 16384, DIM = 1024 fp32 (4 KB/row)
// Pure bandwidth problem (~128 MB total traffic -> ~5.6 us at 23.3 TB/s).
// CDNA5 path: async DMA  global -> LDS -> global  (ASYNCcnt-tracked), so row
// data never passes through VGPRs. Weight table (128 MB) fits in the 192 MB
// L2 -> weight loads keep RT caching; output stores are marked NT (TH=1) so
// the 64 MB write-once stream does not evict the L2-resident table.

#define DIM 1024
#define NROWS (4 * 4096)
#define ROWS_PER_BLOCK 8
#define THREADS 256  // 8 waves of 32; 256 lanes * 16B = 4 KB = one row

#define AS1 __attribute__((address_space(1)))
#define AS3 __attribute__((address_space(3)))

// gfx12 CPol immediate: TH in bits [2:0] (0=RT, 1=NT), scope in bits [4:3].
#define CPOL_RT 0
#define CPOL_NT 1

// The async-LDS builtins take int4 pointers (vector_size(16)) in AS1/AS3.
typedef int v4i __attribute__((vector_size(16)));
typedef float v4f __attribute__((ext_vector_type(4)));

#if __has_builtin(__builtin_amdgcn_global_load_async_to_lds_b128) && \
    __has_builtin(__builtin_amdgcn_global_store_async_from_lds_b128)
#define USE_ASYNC_LDS 1
#else
#define USE_ASYNC_LDS 0
#endif

__device__ __forceinline__ void wait_asynccnt0() {
#if __has_builtin(__builtin_amdgcn_s_wait_asynccnt)
  __builtin_amdgcn_s_wait_asynccnt(0);
#else
  asm volatile("s_wait_asynccnt 0" ::: "memory");
#endif
}

__global__ __launch_bounds__(THREADS) void Embedding_gather_kernel(
    const int* __restrict__ tok, const float* __restrict__ w,
    float* __restrict__ out) {
  const int t = threadIdx.x;            // 0..255, owns 16B chunk t of a row
  const int row0 = blockIdx.x * ROWS_PER_BLOCK;

#if USE_ASYNC_LDS
  __shared__ float smem[ROWS_PER_BLOCK * DIM];  // 32 KB of the 320 KB WGP LDS

  // Phase 1: 8 outstanding async B128 loads per lane, HBM/L2 -> LDS.
  // tok[row0+i] depends only on blockIdx -> uniform -> one scalar s_load_b256.
#pragma unroll
  for (int i = 0; i < ROWS_PER_BLOCK; ++i) {
    const int id = tok[row0 + i];
    const float* src = w + (size_t)id * DIM + t * 4;
    __builtin_amdgcn_global_load_async_to_lds_b128(
        (AS1 v4i*)(void*)src,
        (AS3 v4i*)&smem[i * DIM + t * 4],
        /*imm offset*/ 0, /*cpol*/ CPOL_RT);
  }

  // Each lane stores exactly the LDS bytes its own loads wrote -> per-wave
  // ASYNCcnt wait suffices, no workgroup barrier needed.
  wait_asynccnt0();

  // Phase 2: 8 async B128 stores per lane, LDS -> output stream (NT).
#pragma unroll
  for (int i = 0; i < ROWS_PER_BLOCK; ++i) {
    float* dst = out + (size_t)(row0 + i) * DIM + t * 4;
    __builtin_amdgcn_global_store_async_from_lds_b128(
        (AS1 v4i*)(void*)dst,
        (AS3 v4i*)&smem[i * DIM + t * 4],
        /*imm offset*/ 0, /*cpol*/ CPOL_NT);
  }

  wait_asynccnt0();  // drain before endpgm (endpgm also waits-idle)
#else
  // Fallback: plain coalesced float4 gather; non-temporal stores keep the
  // 64 MB output stream from evicting the L2-resident weight table.
#pragma unroll
  for (int i = 0; i < ROWS_PER_BLOCK; ++i) {
    const int id = tok[row0 + i];
    const v4f v = *(const v4f*)(w + (size_t)id * DIM + t * 4);
    __builtin_nontemporal_store(v, (v4f*)(out + (size_t)(row0 + i) * DIM + t * 4));
  }
#endif
}

extern "C" void kernel_launch(void* const* d_in, const int* in_sizes, int n_in,
                              void* d_out, int out_size, void* d_ws, size_t ws_size,
                              hipStream_t stream) {
  const int* tok = (const int*)d_in[0];   // [4, 4096] int32 token ids
  const float* w = (const float*)d_in[1]; // [32000, 1024] fp32 weight
  float* out = (float*)d_out;             // [4, 4096, 1024] fp32

  dim3 grid(NROWS / ROWS_PER_BLOCK);      // 2048 blocks
  dim3 block(THREADS);                    // 8 waves (wave32)
  Embedding_gather_kernel<<<grid, block, 0, stream>>>(tok, w, out);
}